// HeteroGraphSAGE_29807073034770
// MI455X (gfx1250) — compile-verified
//
#include <hip/hip_runtime.h>
#include <math.h>

typedef __attribute__((ext_vector_type(2))) float v2f;
typedef __attribute__((ext_vector_type(8))) float v8f;

#define N_CARD 100000
#define N_MERCH 50000
#define NEDGE   1000000
#define DIN     64
#define HDIM    128

__device__ __forceinline__ void atomAddF(float* p, float v) {
    __hip_atomic_fetch_add(p, v, __ATOMIC_RELAXED, __HIP_MEMORY_SCOPE_AGENT);
}

// ---------------------------------------------------------------- degrees
__global__ __launch_bounds__(256)
void count_deg_kernel(const int* __restrict__ dst, float* __restrict__ cnt, int E) {
    int e = blockIdx.x * 256 + threadIdx.x;
    if (e < E) atomAddF(&cnt[dst[e]], 1.0f);
}

// ---------------------------------------------------------------- weight transpose: Wt[n][k] = W[k][n]
__global__ __launch_bounds__(256)
void transpose_kernel(const float* __restrict__ W, float* __restrict__ Wt, int K, int N) {
    int i = blockIdx.x * 256 + threadIdx.x;
    if (i >= K * N) return;
    int k = i / N, n = i - k * N;
    Wt[(size_t)n * K + k] = W[i];
}

// ---------------------------------------------------------------- scatter-add of source rows by destination
// One edge handled by D/4 threads; float4 gather + 4 fp32 global atomics.
template<int D>
__global__ __launch_bounds__(256)
void scatter_add_kernel(const float* __restrict__ x, const int* __restrict__ src,
                        const int* __restrict__ dst, float* __restrict__ agg, int E) {
    constexpr int TPE = D / 4;        // threads per edge (16 or 32)
    constexpr int EPB = 256 / TPE;    // edges per block
    int e = blockIdx.x * EPB + (int)(threadIdx.x / TPE);
    if (e >= E) return;
    int q = (threadIdx.x % TPE) * 4;
    int s = src[e];
    int d = dst[e];
    const float4 v = *reinterpret_cast<const float4*>(x + (size_t)s * D + q);
    float* a = agg + (size_t)d * D + q;
    atomAddF(a + 0, v.x);
    atomAddF(a + 1, v.y);
    atomAddF(a + 2, v.z);
    atomAddF(a + 3, v.w);
}

// ---------------------------------------------------------------- fused SAGE layer:
// out[16 x 128] = relu( (agg @ Wl) * rinv_row + bl + xdst @ Wr ), one 16-node tile per block.
// 8 waves; wave w computes output columns [16w, 16w+16) with V_WMMA_F32_16X16X4_F32.
// Wlt / Wrt are pre-transposed [N=128][K] so each lane's (k0,k0+1) B pair is one b64 load.
template<int K>
__global__ __launch_bounds__(256)
void sage_gemm_kernel(const float* __restrict__ agg, const float* __restrict__ cnt,
                      const float* __restrict__ xdst,
                      const float* __restrict__ Wlt, const float* __restrict__ bl,
                      const float* __restrict__ Wrt, float* __restrict__ out) {
    __shared__ float sAgg[16][K + 4];   // +4 pad: conflict-free column reads
    __shared__ float sX[16][K + 4];
    __shared__ float sRinv[16];
    const int row0 = blockIdx.x * 16;
    const int tid  = threadIdx.x;

    for (int i = tid; i < 16 * K; i += 256) {
        int r = i / K, k = i - r * K;
        int node = row0 + r;
        sAgg[r][k] = agg[(size_t)node * K + k];
        sX[r][k]   = xdst[(size_t)node * K + k];
    }
    if (tid < 16) {
        float c = cnt[row0 + tid];
        sRinv[tid] = 1.0f / (c > 1.0f ? c : 1.0f);
    }
    __syncthreads();

    const int wave = tid >> 5;
    const int lane = tid & 31;
    const int half = lane >> 4;   // selects K pair within 4-chunk
    const int ln   = lane & 15;   // M for A-reads, N for B/C/D
    const int col  = wave * 16 + ln;

    v8f acc = {};
#pragma unroll
    for (int kk = 0; kk < K; kk += 4) {
        int k0 = kk + 2 * half;
        v2f a;
        a.x = sAgg[ln][k0];
        a.y = sAgg[ln][k0 + 1];
        const v2f b = *reinterpret_cast<const v2f*>(Wlt + (size_t)col * K + k0);
        acc = __builtin_amdgcn_wmma_f32_16x16x4_f32(false, a, false, b, (short)0, acc, false, false);
    }
    // mean normalization commutes with the right-multiply: scale accumulator rows
#pragma unroll
    for (int j = 0; j < 8; ++j) acc[j] *= sRinv[j + 8 * half];
#pragma unroll
    for (int kk = 0; kk < K; kk += 4) {
        int k0 = kk + 2 * half;
        v2f a;
        a.x = sX[ln][k0];
        a.y = sX[ln][k0 + 1];
        const v2f b = *reinterpret_cast<const v2f*>(Wrt + (size_t)col * K + k0);
        acc = __builtin_amdgcn_wmma_f32_16x16x4_f32(false, a, false, b, (short)0, acc, false, false);
    }

    const float bias = bl[col];
#pragma unroll
    for (int j = 0; j < 8; ++j) {
        int m = j + 8 * half;    // C/D layout: vgpr j -> row j (lanes 0-15) / j+8 (lanes 16-31)
        float v = acc[j] + bias;
        v = v > 0.0f ? v : 0.0f;
        out[(size_t)(row0 + m) * HDIM + col] = v;
    }
}

// ---------------------------------------------------------------- heads: classifier (128->64->2) + risk (128->64->1, sigmoid)
// cW1t / rW1t pre-transposed [64][128].
__global__ __launch_bounds__(256)
void head_kernel(const float* __restrict__ h,
                 const float* __restrict__ cW1t, const float* __restrict__ cb1,
                 const float* __restrict__ cW2, const float* __restrict__ cb2,
                 const float* __restrict__ rW1t, const float* __restrict__ rb1,
                 const float* __restrict__ rW2, const float* __restrict__ rb2,
                 float* __restrict__ pred, float* __restrict__ risk) {
    __shared__ float sH[16][HDIM + 4];
    __shared__ float sHid[16][68];
    __shared__ float sRisk[16][68];
    const int row0 = blockIdx.x * 16;
    const int tid  = threadIdx.x;

    for (int i = tid; i < 16 * HDIM; i += 256) {
        int r = i >> 7, k = i & 127;
        sH[r][k] = h[(size_t)(row0 + r) * HDIM + k];
    }
    __syncthreads();

    const int wave = tid >> 5;
    const int lane = tid & 31;
    const int half = lane >> 4;
    const int ln   = lane & 15;
    const int w    = wave & 3;
    const int col  = w * 16 + ln;          // 0..63 hidden unit
    const bool isRisk = wave >= 4;
    const float* W1t = isRisk ? rW1t : cW1t;
    const float* b1  = isRisk ? rb1 : cb1;

    v8f acc = {};
#pragma unroll
    for (int kk = 0; kk < HDIM; kk += 4) {
        int k0 = kk + 2 * half;
        v2f a;
        a.x = sH[ln][k0];
        a.y = sH[ln][k0 + 1];
        const v2f b = *reinterpret_cast<const v2f*>(W1t + (size_t)col * HDIM + k0);
        acc = __builtin_amdgcn_wmma_f32_16x16x4_f32(false, a, false, b, (short)0, acc, false, false);
    }

    const float bias = b1[col];
    float (*dstLds)[68] = isRisk ? sRisk : sHid;
#pragma unroll
    for (int j = 0; j < 8; ++j) {
        int m = j + 8 * half;
        float v = acc[j] + bias;
        dstLds[m][col] = v > 0.0f ? v : 0.0f;
    }
    __syncthreads();

    if (tid < 16) {
        int node = row0 + tid;
        float p0 = cb2[0], p1 = cb2[1], rr = rb2[0];
#pragma unroll 4
        for (int k = 0; k < 64; ++k) {
            float z = sHid[tid][k];
            p0 += z * cW2[k * 2 + 0];
            p1 += z * cW2[k * 2 + 1];
            rr += sRisk[tid][k] * rW2[k];
        }
        pred[(size_t)node * 2 + 0] = p0;
        pred[(size_t)node * 2 + 1] = p1;
        risk[node] = 1.0f / (1.0f + __expf(-rr));
    }
}

// ---------------------------------------------------------------- driver
extern "C" void kernel_launch(void* const* d_in, const int* in_sizes, int n_in,
                              void* d_out, int out_size, void* d_ws, size_t ws_size,
                              hipStream_t stream) {
    const float* x_card  = (const float*)d_in[0];
    const float* x_merch = (const float*)d_in[1];
    const int* src_cm = (const int*)d_in[2];
    const int* dst_cm = (const int*)d_in[3];
    const int* src_mc = (const int*)d_in[4];
    const int* dst_mc = (const int*)d_in[5];
    const float *Wl0_cm = (const float*)d_in[6],  *bl0_cm = (const float*)d_in[7],  *Wr0_cm = (const float*)d_in[8];
    const float *Wl0_mc = (const float*)d_in[9],  *bl0_mc = (const float*)d_in[10], *Wr0_mc = (const float*)d_in[11];
    const float *Wl1_cm = (const float*)d_in[12], *bl1_cm = (const float*)d_in[13], *Wr1_cm = (const float*)d_in[14];
    const float *Wl1_mc = (const float*)d_in[15], *bl1_mc = (const float*)d_in[16], *Wr1_mc = (const float*)d_in[17];
    const float *cW1_card  = (const float*)d_in[18], *cb1_card  = (const float*)d_in[19];
    const float *cW2_card  = (const float*)d_in[20], *cb2_card  = (const float*)d_in[21];
    const float *cW1_merch = (const float*)d_in[22], *cb1_merch = (const float*)d_in[23];
    const float *cW2_merch = (const float*)d_in[24], *cb2_merch = (const float*)d_in[25];
    const float *rW1 = (const float*)d_in[26], *rb1 = (const float*)d_in[27];
    const float *rW2 = (const float*)d_in[28], *rb2 = (const float*)d_in[29];

    // outputs (concatenated flat, reference return order)
    float* out    = (float*)d_out;
    float* h_c2   = out;
    float* h_m2   = h_c2 + (size_t)N_CARD * HDIM;
    float* pred_c = h_m2 + (size_t)N_MERCH * HDIM;
    float* pred_m = pred_c + (size_t)N_CARD * 2;
    float* risk_c = pred_m + (size_t)N_MERCH * 2;
    float* risk_m = risk_c + N_CARD;

    // workspace layout
    float* ws    = (float*)d_ws;
    float* h_c   = ws;                                // 100000*128
    float* h_m   = h_c + (size_t)N_CARD * HDIM;       // 50000*128
    float* agg   = h_m + (size_t)N_MERCH * HDIM;      // 100000*128 (reused per phase)
    float* cnt_c = agg + (size_t)N_CARD * HDIM;       // 100000
    float* cnt_m = cnt_c + N_CARD;                    // 50000
    // transposed weights
    float* p = cnt_m + N_MERCH;
    float* tWl0_cm = p; p += DIN * HDIM;
    float* tWr0_cm = p; p += DIN * HDIM;
    float* tWl0_mc = p; p += DIN * HDIM;
    float* tWr0_mc = p; p += DIN * HDIM;
    float* tWl1_cm = p; p += HDIM * HDIM;
    float* tWr1_cm = p; p += HDIM * HDIM;
    float* tWl1_mc = p; p += HDIM * HDIM;
    float* tWr1_mc = p; p += HDIM * HDIM;
    float* tcW1_card  = p; p += HDIM * 64;
    float* tcW1_merch = p; p += HDIM * 64;
    float* trW1       = p; p += HDIM * 64;

    const int ebl = (NEDGE + 255) / 256;
    const int sgrid64  = (NEDGE + (256 / (DIN / 4)) - 1) / (256 / (DIN / 4));
    const int sgrid128 = (NEDGE + (256 / (HDIM / 4)) - 1) / (256 / (HDIM / 4));
    const int t64  = (DIN * HDIM + 255) / 256;
    const int t128 = (HDIM * HDIM + 255) / 256;
    const int th   = (HDIM * 64 + 255) / 256;

    // weight transposes (independent of everything else)
    transpose_kernel<<<t64, 256, 0, stream>>>(Wl0_cm, tWl0_cm, DIN, HDIM);
    transpose_kernel<<<t64, 256, 0, stream>>>(Wr0_cm, tWr0_cm, DIN, HDIM);
    transpose_kernel<<<t64, 256, 0, stream>>>(Wl0_mc, tWl0_mc, DIN, HDIM);
    transpose_kernel<<<t64, 256, 0, stream>>>(Wr0_mc, tWr0_mc, DIN, HDIM);
    transpose_kernel<<<t128, 256, 0, stream>>>(Wl1_cm, tWl1_cm, HDIM, HDIM);
    transpose_kernel<<<t128, 256, 0, stream>>>(Wr1_cm, tWr1_cm, HDIM, HDIM);
    transpose_kernel<<<t128, 256, 0, stream>>>(Wl1_mc, tWl1_mc, HDIM, HDIM);
    transpose_kernel<<<t128, 256, 0, stream>>>(Wr1_mc, tWr1_mc, HDIM, HDIM);
    transpose_kernel<<<th, 256, 0, stream>>>(cW1_card,  tcW1_card,  HDIM, 64);
    transpose_kernel<<<th, 256, 0, stream>>>(cW1_merch, tcW1_merch, HDIM, 64);
    transpose_kernel<<<th, 256, 0, stream>>>(rW1,       trW1,       HDIM, 64);

    // degrees (edge lists identical for both layers)
    hipMemsetAsync(cnt_c, 0, (size_t)N_CARD * sizeof(float), stream);
    hipMemsetAsync(cnt_m, 0, (size_t)N_MERCH * sizeof(float), stream);
    count_deg_kernel<<<ebl, 256, 0, stream>>>(dst_cm, cnt_m, NEDGE);
    count_deg_kernel<<<ebl, 256, 0, stream>>>(dst_mc, cnt_c, NEDGE);

    // ---- layer 0: merchants  (msgs from cards over cm edges)
    hipMemsetAsync(agg, 0, (size_t)N_MERCH * DIN * sizeof(float), stream);
    scatter_add_kernel<DIN><<<sgrid64, 256, 0, stream>>>(x_card, src_cm, dst_cm, agg, NEDGE);
    sage_gemm_kernel<DIN><<<N_MERCH / 16, 256, 0, stream>>>(agg, cnt_m, x_merch, tWl0_cm, bl0_cm, tWr0_cm, h_m);

    // ---- layer 0: cards  (msgs from merchants over mc edges)
    hipMemsetAsync(agg, 0, (size_t)N_CARD * DIN * sizeof(float), stream);
    scatter_add_kernel<DIN><<<sgrid64, 256, 0, stream>>>(x_merch, src_mc, dst_mc, agg, NEDGE);
    sage_gemm_kernel<DIN><<<N_CARD / 16, 256, 0, stream>>>(agg, cnt_c, x_card, tWl0_mc, bl0_mc, tWr0_mc, h_c);

    // ---- layer 1: merchants (msgs = h_c over cm edges)
    hipMemsetAsync(agg, 0, (size_t)N_MERCH * HDIM * sizeof(float), stream);
    scatter_add_kernel<HDIM><<<sgrid128, 256, 0, stream>>>(h_c, src_cm, dst_cm, agg, NEDGE);
    sage_gemm_kernel<HDIM><<<N_MERCH / 16, 256, 0, stream>>>(agg, cnt_m, h_m, tWl1_cm, bl1_cm, tWr1_cm, h_m2);

    // ---- layer 1: cards (msgs = h_m over mc edges)
    hipMemsetAsync(agg, 0, (size_t)N_CARD * HDIM * sizeof(float), stream);
    scatter_add_kernel<HDIM><<<sgrid128, 256, 0, stream>>>(h_m, src_mc, dst_mc, agg, NEDGE);
    sage_gemm_kernel<HDIM><<<N_CARD / 16, 256, 0, stream>>>(agg, cnt_c, h_c, tWl1_mc, bl1_mc, tWr1_mc, h_c2);

    // ---- heads
    head_kernel<<<N_CARD / 16, 256, 0, stream>>>(h_c2, tcW1_card, cb1_card, cW2_card, cb2_card,
                                                 trW1, rb1, rW2, rb2, pred_c, risk_c);
    head_kernel<<<N_MERCH / 16, 256, 0, stream>>>(h_m2, tcW1_merch, cb1_merch, cW2_merch, cb2_merch,
                                                  trW1, rb1, rW2, rb2, pred_m, risk_m);
}